// PoissonNeuronTransform_1726576857126
// MI455X (gfx1250) — compile-verified
//
#include <hip/hip_runtime.h>
#include <math.h>
#include <stdint.h>

// ---------------------------------------------------------------------------
// Poisson spike counting, bit-matched to JAX threefry2x32 (key = (0,42)).
// Shape: (neurons=256, trials=32, timesteps=128) -> 1,048,576 elements.
// deltas shape (100, 1048576): flat counter idx = s*2^20 + e.
// JAX splits counters in half (half = 50*2^20), so threefry((s*2^20+e,
// (s+50)*2^20+e)) yields bits for spike s (lane0) AND spike s+50 (lane1).
// ---------------------------------------------------------------------------

#define THREADS      256
#define ELEMS        1048576u      // 256*32*128
#define HALF_SPIKES  50
#define REFRACTORY   0.002f
#define T_INTERVAL   1.0f

// LDS layout (dynamic shared, starts at LDS byte 0):
//   [0 , 51200)      : per-thread staging of lane-1 threefry outputs
//   [51200 , 53248)  : TDM landing pad for this block's 256 rates (+pad)
#define LDS_O1_BYTES   (HALF_SPIKES * THREADS * 4)   // 51200
#define LDS_TDM_ADDR   51200u
#define LDS_TOTAL      53248u

typedef unsigned int v4u __attribute__((ext_vector_type(4)));
typedef int          v8i __attribute__((ext_vector_type(8)));
typedef int          v4i __attribute__((ext_vector_type(4)));

__device__ __forceinline__ uint32_t rotl32(uint32_t x, uint32_t d) {
  return (x << d) | (x >> (32u - d));
}

__device__ __forceinline__ void tf_round(uint32_t &x0, uint32_t &x1, uint32_t d) {
  x0 += x1;
  x1 = rotl32(x1, d);
  x1 ^= x0;
}

// Threefry-2x32 with fixed key (0, 42) — matches jax.random.key(42).
__device__ __forceinline__ void threefry_0_42(uint32_t c0, uint32_t c1,
                                              uint32_t &o0, uint32_t &o1) {
  const uint32_t k0 = 0u;
  const uint32_t k1 = 42u;
  const uint32_t k2 = 0x1BD11BDAu ^ k0 ^ k1;
  uint32_t x0 = c0 + k0;
  uint32_t x1 = c1 + k1;
  tf_round(x0, x1, 13); tf_round(x0, x1, 15); tf_round(x0, x1, 26); tf_round(x0, x1, 6);
  x0 += k1; x1 += k2 + 1u;
  tf_round(x0, x1, 17); tf_round(x0, x1, 29); tf_round(x0, x1, 16); tf_round(x0, x1, 24);
  x0 += k2; x1 += k0 + 2u;
  tf_round(x0, x1, 13); tf_round(x0, x1, 15); tf_round(x0, x1, 26); tf_round(x0, x1, 6);
  x0 += k0; x1 += k1 + 3u;
  tf_round(x0, x1, 17); tf_round(x0, x1, 29); tf_round(x0, x1, 16); tf_round(x0, x1, 24);
  x0 += k1; x1 += k2 + 4u;
  tf_round(x0, x1, 13); tf_round(x0, x1, 15); tf_round(x0, x1, 26); tf_round(x0, x1, 6);
  x0 += k2; x1 += k0 + 5u;
  o0 = x0; o1 = x1;
}

// bits -> f in [1,2) -> u = f-1 in [0,1) -> Exp(1) sample -log1p(-u).
// (f-1) and (1-f) are Sterbenz-exact, so this matches JAX's formulation.
__device__ __forceinline__ float exp_sample(uint32_t bits) {
  const float f = __uint_as_float((bits >> 9) | 0x3F800000u);
  return -log1pf(1.0f - f);
}

__global__ __launch_bounds__(THREADS)
void poisson_count_kernel(const float* __restrict__ rates,
                          float* __restrict__ out) {
  extern __shared__ uint32_t smem[];
  uint32_t* s_o1 = smem;  // s_o1[s*THREADS + tid], private per-thread staging

  const uint32_t tid = threadIdx.x;
  const uint32_t e   = (uint32_t)blockIdx.x * THREADS + tid;

  // -------------------------------------------------------------------------
  // CDNA5 Tensor Data Mover: one wave async-stages this block's 256 rates
  // (1 KB, 1-D tile) into an LDS landing pad. This drags the lines through
  // the cache hierarchy ahead of the per-lane global load below. D# built
  // per ISA ch.8: group0 {count=1, lds_addr, global_addr, type=2};
  // group1 {data_size=4B, tensor_dim0=256, tile_dim0=256, stride0=256}.
  // -------------------------------------------------------------------------
  if (tid < 32u) {
    const uint64_t ga = (uint64_t)(uintptr_t)(rates + (size_t)blockIdx.x * THREADS);
    v4u g0;
    g0.x = 1u;                                                  // count=1 (valid)
    g0.y = LDS_TDM_ADDR;                                        // lds_addr bytes
    g0.z = (uint32_t)ga;                                        // global_addr lo
    g0.w = ((uint32_t)(ga >> 32) & 0x01FFFFFFu) | (2u << 30);   // addr[56:32]|type=2
    v8i g1;
    g1[0] = 0x00020000;          // data_size=2 -> 4-byte elements
    g1[1] = (int)(256u << 16);   // tensor_dim0 = 256 (bits 79:48, low half)
    g1[2] = (int)(1u << 16);     // tensor_dim1 = 1
    g1[3] = (int)(256u << 16);   // tile_dim0 = 256 (bits 127:112)
    g1[4] = 1;                   // tile_dim1 = 1, tile_dim2 = 0 (unused)
    g1[5] = 256;                 // tensor_dim0_stride = 256
    g1[6] = 0;
    g1[7] = 0;
    v4i g2; g2[0] = 1; g2[1] = 1; g2[2] = 0; g2[3] = 0;         // dim2=dim3=1
    v4i g3; g3[0] = 0; g3[1] = (int)(1u << 16); g3[2] = 0; g3[3] = 0;  // dim4=1
#if defined(__clang_major__) && (__clang_major__ >= 23)
    v8i g4 = {0, 0, 0, 0, 0, 0, 0, 0};
    __builtin_amdgcn_tensor_load_to_lds(g0, g1, g2, g3, g4, 0);
#else
    __builtin_amdgcn_tensor_load_to_lds(g0, g1, g2, g3, 0);
#endif
    __builtin_amdgcn_s_wait_tensorcnt(0);
  }
  __syncthreads();

  const float rate = rates[e];   // hits L2/WGP$ (TDM already pulled the lines)

  float t     = 0.0f;
  int   count = 0;
  bool  alive = true;

  // ---- spikes 0..49: each threefry yields bits for s (use now) and s+50
  // (stash in LDS). Early exit once every lane in the wave32 passed 1.0 s —
  // deltas are strictly positive so counts are final.
  uint32_t c0 = e;
  #pragma unroll 2
  for (int s = 0; s < HALF_SPIKES; ++s, c0 += ELEMS) {
    uint32_t o0, o1;
    threefry_0_42(c0, c0 + (uint32_t)HALF_SPIKES * ELEMS, o0, o1);
    s_o1[(uint32_t)s * THREADS + tid] = o1;
    t += exp_sample(o0) / rate + REFRACTORY;
    count += (t < T_INTERVAL) ? 1 : 0;
    if (__all(t >= T_INTERVAL)) { alive = false; break; }
  }

  // ---- spikes 50..99: replay the stashed lane-1 bits from LDS (no threefry
  // recompute). Only reached if the first loop ran to completion, so all 50
  // staged values are valid. Each thread reads only its own column.
  if (alive) {
    #pragma unroll 2
    for (int s = 0; s < HALF_SPIKES; ++s) {
      const uint32_t o1 = s_o1[(uint32_t)s * THREADS + tid];
      t += exp_sample(o1) / rate + REFRACTORY;
      count += (t < T_INTERVAL) ? 1 : 0;
      if (__all(t >= T_INTERVAL)) break;
    }
  }

  out[e] = (float)count;   // int32 counts, exactly representable in f32
}

extern "C" void kernel_launch(void* const* d_in, const int* in_sizes, int n_in,
                              void* d_out, int out_size, void* d_ws, size_t ws_size,
                              hipStream_t stream) {
  (void)in_sizes; (void)n_in; (void)d_ws; (void)ws_size;
  const float* rates = (const float*)d_in[0];
  float* out = (float*)d_out;
  const int nblocks = (out_size + THREADS - 1) / THREADS;  // 4096
  poisson_count_kernel<<<nblocks, THREADS, LDS_TOTAL, stream>>>(rates, out);
}